// AdaptiveGridMerger_80264348828010
// MI455X (gfx1250) — compile-verified
//
#include <hip/hip_runtime.h>
#include <hip/hip_bf16.h>

typedef __attribute__((ext_vector_type(2))) float v2f;
typedef __attribute__((ext_vector_type(8))) float v8f;

#define BB 16
#define CC 306
#define TT 4096
#define MM 270
#define GG 64
#define TILE_T 128   // t columns per block
#define NTHREADS 256 // 8 waves of 32

// Fused kernel: bilinear scatter into LDS grid tile, then W @ grid via f32 WMMA.
// grid = (TT/TILE_T, BB), block = 256 threads (8 wave32).
__global__ __launch_bounds__(NTHREADS)
void AdaptiveGridMerger_fused_kernel(const float* __restrict__ x,
                                     const float* __restrict__ positions,
                                     const float* __restrict__ W,
                                     float* __restrict__ out) {
    __shared__ float gv[GG * TILE_T];   // 32 KB grid tile [g][t]
    __shared__ int   tg[CC * 4];        // 4 corner flat-grid indices per channel
    __shared__ float tw[CC * 4];        // 4 corner weights per channel

    const int b  = blockIdx.y;
    const int t0 = blockIdx.x * TILE_T;
    const int tid = threadIdx.x;

    // ---- Phase 0: per-channel bilinear corner table (uniform over t) ----
    for (int c = tid; c < CC; c += NTHREADS) {
        const float px = positions[((size_t)b * CC + c) * 2 + 0];
        const float py = positions[((size_t)b * CC + c) * 2 + 1];
        const float g0 = (px + 1.0f) * 4.0f;   // (p+1)*grid_size/2, grid_size=8
        const float g1 = (py + 1.0f) * 4.0f;
        const float l0 = floorf(g0), l1 = floorf(g1);
        const int i0l = (int)l0,          i1l = (int)l1;
        const int i0h = (int)ceilf(g0),   i1h = (int)ceilf(g1);
        const float wh0 = g0 - l0, wl0 = 1.0f - wh0;
        const float wh1 = g1 - l1, wl1 = 1.0f - wh1;
        tg[c * 4 + 0] = i0l * 8 + i1l;  tw[c * 4 + 0] = wl0 * wl1;
        tg[c * 4 + 1] = i0l * 8 + i1h;  tw[c * 4 + 1] = wl0 * wh1;
        tg[c * 4 + 2] = i0h * 8 + i1l;  tw[c * 4 + 2] = wh0 * wl1;
        tg[c * 4 + 3] = i0h * 8 + i1h;  tw[c * 4 + 3] = wh0 * wh1;
    }
    // zero the grid tile
    for (int i = tid; i < GG * TILE_T; i += NTHREADS) gv[i] = 0.0f;
    __syncthreads();

    // ---- Phase 1: scatter-add x into LDS grid tile (ds_add_f32, no-return) ----
    {
        const int t  = tid & (TILE_T - 1);      // lane owns one t column
        const int ch = tid >> 7;                // split C across two thread halves
        const int cbeg = ch * (CC / 2);
        const int cend = cbeg + (CC / 2);
        const float* xb = x + ((size_t)b * CC) * TT + t0 + t;
        for (int c = cbeg; c < cend; ++c) {
            const float xv = xb[(size_t)c * TT];
            #pragma unroll
            for (int j = 0; j < 4; ++j) {
                atomicAdd(&gv[tg[c * 4 + j] * TILE_T + t], tw[c * 4 + j] * xv);
            }
        }
    }
    __syncthreads();

    // ---- Phase 2: out[b, :, t-tile] = W(270x64) @ gv(64x128) via WMMA f32 ----
    const int lane = tid & 31;
    const int wv   = tid >> 5;        // wave id 0..7 -> 16-wide n sub-tile
    const int lo   = lane & 15;
    const int hi   = lane >> 4;       // k-half / m-half selector
    const int nloc = wv * 16 + lo;    // column within the 128-wide tile

    // Preload all B fragments (m-tile invariant): 16 K-steps x 2 VGPRs.
    float bxk[16], byk[16];
    #pragma unroll
    for (int kk = 0; kk < 16; ++kk) {
        const int kb = kk * 4 + hi * 2;
        bxk[kk] = gv[kb * TILE_T + nloc];
        byk[kk] = gv[(kb + 1) * TILE_T + nloc];
    }

    float* outb = out + ((size_t)b * MM) * TT + t0 + wv * 16 + lo;

    for (int mt = 0; mt < 17; ++mt) {         // ceil(270/16) m-tiles
        const int m0 = mt * 16;
        int ma = m0 + lo;                     // A-fragment row; clamp pad rows
        if (ma > MM - 1) ma = MM - 1;
        v8f acc = {};
        #pragma unroll
        for (int kk = 0; kk < 16; ++kk) {
            const int kb = kk * 4 + hi * 2;
            v2f a = *(const v2f*)(W + (size_t)ma * GG + kb);   // A 16x4 fragment
            v2f bfrag;
            bfrag.x = bxk[kk];
            bfrag.y = byk[kk];
            acc = __builtin_amdgcn_wmma_f32_16x16x4_f32(
                /*neg_a=*/false, a, /*neg_b=*/false, bfrag,
                /*c_mod=*/(short)0, acc, /*reuse_a=*/false, /*reuse_b=*/false);
        }
        #pragma unroll
        for (int i = 0; i < 8; ++i) {
            const int m = m0 + hi * 8 + i;    // C/D layout: VGPR i -> M = i + 8*hi
            if (m < MM) outb[(size_t)m * TT] = acc[i];
        }
    }
}

extern "C" void kernel_launch(void* const* d_in, const int* in_sizes, int n_in,
                              void* d_out, int out_size, void* d_ws, size_t ws_size,
                              hipStream_t stream) {
    (void)in_sizes; (void)n_in; (void)out_size; (void)d_ws; (void)ws_size;
    const float* x   = (const float*)d_in[0];   // [B, C, T] f32
    const float* pos = (const float*)d_in[1];   // [B, C, 2] f32
    const float* W   = (const float*)d_in[2];   // [M, G] f32
    float* out = (float*)d_out;                 // [B, M, T] f32

    dim3 grid(TT / TILE_T, BB);
    dim3 block(NTHREADS);
    AdaptiveGridMerger_fused_kernel<<<grid, block, 0, stream>>>(x, pos, W, out);
}